// DecoderRNN_52725018526321
// MI455X (gfx1250) — compile-verified
//
#include <hip/hip_runtime.h>
#include <cmath>

// Problem dims (fixed by the reference)
constexpr int B  = 64;      // batch
constexpr int T  = 32;      // caption length
constexpr int S  = 33;      // T + 1 (features prepended)
constexpr int V  = 32000;   // vocab
constexpr int E  = 512;     // embed
constexpr int H  = 1024;    // hidden
constexpr int G  = 4 * H;   // gates (i,f,g,o)

constexpr int KC  = 64;     // k-chunk staged in LDS per TDM transfer
constexpr int NKC = H / KC; // 16 chunks per step
constexpr int LPAD = 4;     // LDS row pad (floats) -> banks 4*row+k, conflict-free

typedef __attribute__((ext_vector_type(2))) float v2f;
typedef __attribute__((ext_vector_type(8))) float v8f;
typedef __attribute__((ext_vector_type(4))) unsigned int v4u;
typedef __attribute__((ext_vector_type(4))) int v4i;
typedef __attribute__((ext_vector_type(8))) int v8i;

__device__ inline float sigmoidf_(float x) { return 1.0f / (1.0f + __expf(-x)); }

__device__ inline unsigned lds_off(const void* p) {
  // Generic pointer to LDS: low 32 bits are the LDS byte offset (aperture doc).
  return (unsigned)(uintptr_t)p;
}

// ---------------------------------------------------------------------------
// TDM: DMA a [64 rows x 64 cols] fp32 chunk of h (row-major, ld=H) into LDS,
// padding each 64-DWORD row with 4 DWORDs so LDS rows are KC+LPAD floats.
// D# per CDNA5 ISA. Tracked by TENSORcnt.
// ---------------------------------------------------------------------------
__device__ inline void tdm_load_h_chunk(const float* h_in, int k0, unsigned lds_addr) {
  unsigned long long ga = (unsigned long long)(uintptr_t)(h_in + k0);
  v4u g0;
  g0[0] = 1u;                                   // count=1, user descriptor
  g0[1] = lds_addr;                             // LDS byte address
  g0[2] = (unsigned)(ga & 0xffffffffu);         // global_addr[31:0]
  g0[3] = (unsigned)((ga >> 32) & 0x01ffffffu)  // global_addr[56:32]
        | (2u << 30);                           // type=2 ("image")
  v8i g1;
  g1[0] = (int)((2u << 16)                      // data_size = 4 bytes
              | (1u << 20)                      // pad_enable
              | (5u << 22)                      // pad_interval: 64 DWORDs
              | (3u << 25));                    // pad_amount: 4 DWORDs
  g1[1] = (int)((unsigned)H << 16);             // tensor_dim0[15:0] = 1024
  g1[2] = (int)((unsigned)B << 16);             // tensor_dim0 hi=0; tensor_dim1 = 64
  g1[3] = (int)((unsigned)KC << 16);            // tensor_dim1 hi=0; tile_dim0 = 64
  g1[4] = B;                                    // tile_dim1 = 64; tile_dim2 = 0 (2D)
  g1[5] = H;                                    // tensor_dim0_stride lo = 1024
  g1[6] = 0;                                    // stride hi; tensor_dim1_stride lo
  g1[7] = 0;
  v4i gz = {0, 0, 0, 0};                        // groups 2/3 unused (<=2D)
#if __clang_major__ >= 23
  v8i gz8 = {0, 0, 0, 0, 0, 0, 0, 0};
  __builtin_amdgcn_tensor_load_to_lds(g0, g1, gz, gz, gz8, 0);
#else
  __builtin_amdgcn_tensor_load_to_lds(g0, g1, gz, gz, 0);
#endif
}

// ---------------------------------------------------------------------------
// fp32 WMMA helper: one 16x16x4 step.
// ---------------------------------------------------------------------------
__device__ inline v8f wmma4(v2f a, v2f b, v8f acc) {
  return __builtin_amdgcn_wmma_f32_16x16x4_f32(
      false, a, false, b, (short)0, acc, false, false);
}

// ---------------------------------------------------------------------------
// Kernel A: build time-major inputs X[s][b][e]  (s=0: features; s>=1: gather)
// and initialize h (double buffer slot 0) and c from h0/c0.
// ---------------------------------------------------------------------------
__global__ void build_inputs_kernel(const float* __restrict__ features,
                                    const int*   __restrict__ captions,
                                    const float* __restrict__ W_embed,
                                    const float* __restrict__ b_embed,
                                    const float* __restrict__ h0,
                                    const float* __restrict__ c0,
                                    float* __restrict__ X,
                                    float* __restrict__ hbuf0,
                                    float* __restrict__ cbuf) {
  const int NX = S * B * E;
  const int NH = B * H;
  int idx = blockIdx.x * blockDim.x + threadIdx.x;
  if (idx < NX) {
    int s = idx / (B * E);
    int r = idx - s * (B * E);
    int b = r / E;
    int e = r - b * E;
    float v;
    if (s == 0) {
      v = features[b * E + e];
    } else {
      int tok = captions[b * T + (s - 1)];
      v = W_embed[(size_t)e * V + tok] + b_embed[e];
    }
    X[(size_t)(s * B + b) * E + e] = v;
  } else if (idx < NX + NH) {
    int j = idx - NX;
    hbuf0[j] = h0[j];
  } else if (idx < NX + 2 * NH) {
    int j = idx - NX - NH;
    cbuf[j] = c0[j];
  }
}

// ---------------------------------------------------------------------------
// Kernel B: Xg[m][n] = X[m][:] . w_ih[n][:] + b_ih[n] + b_hh[n]
//   M = S*B = 2112 rows, N = G = 4096, K = E = 512
// One wave per 16x16 tile; 8 waves per block.  Per CDNA5 ISA 7.12.2:
//   A (16x4 f32): lanes 0-15 K={k,k+1}, lanes 16-31 K={k+2,k+3}
//   B mirrored by transpose; C/D: VGPR r <-> M = r + 8*half, N = lane&15.
// ---------------------------------------------------------------------------
__global__ void __launch_bounds__(256)
input_gemm_kernel(const float* __restrict__ X,
                  const float* __restrict__ w_ih,
                  const float* __restrict__ b_ih,
                  const float* __restrict__ b_hh,
                  float* __restrict__ Xg) {
  const int NT_N = G / 16;        // 256
  int wave = blockIdx.x * 8 + (threadIdx.x >> 5);
  if (wave >= ((S * B) / 16) * NT_N) return;
  int mt = wave / NT_N;
  int nt = wave - mt * NT_N;

  const int lane = threadIdx.x & 31;
  const int l    = lane & 15;
  const int half = lane >> 4;

  const float2* Arow = (const float2*)(X    + (size_t)(mt * 16 + l) * E);
  const float2* Brow = (const float2*)(w_ih + (size_t)(nt * 16 + l) * E);
  __builtin_prefetch(w_ih + (size_t)(nt * 16) * E, 0, 3);

  v8f acc = {};
#pragma unroll 8
  for (int k = 0; k < E; k += 4) {
    float2 af = Arow[(k >> 1) + half];
    float2 bf = Brow[(k >> 1) + half];
    v2f a; a.x = af.x; a.y = af.y;
    v2f b; b.x = bf.x; b.y = bf.y;
    acc = wmma4(a, b, acc);
  }

  const int n = nt * 16 + l;
  const float bias = b_ih[n] + b_hh[n];
#pragma unroll
  for (int r = 0; r < 8; ++r) {
    int m = mt * 16 + r + half * 8;
    Xg[(size_t)m * G + n] = acc[r] + bias;
  }
}

// ---------------------------------------------------------------------------
// Kernel C (launched once per time step): fused recurrent step.
//   gates[b, gate*H+j] = Xg[s*B+b, gate*H+j] + h_in[b,:] . w_hh[gate*H+j, :]
// Block: 256 threads (8 waves), one block per 16 hidden columns (64 blocks).
// h is DMA'd into LDS in double-buffered 64x64 chunks by the TDM (wave 0
// issues, TENSORcnt-waited, block-barriered); all waves read A-fragments
// from LDS (ds_load_b64, conflict-free via +4 row pad) and B-fragments from
// L2-resident w_hh.  Wave w: gate = w>>1, row tiles {2*(w&1), 2*(w&1)+1}.
// Gate tiles staged in LDS; pointwise LSTM epilogue updates c (in place),
// h_out (double buffer) and the [B,S,H] output.
// ---------------------------------------------------------------------------
__global__ void __launch_bounds__(256)
lstm_step_kernel(const float* __restrict__ Xg,
                 const float* __restrict__ w_hh,
                 const float* __restrict__ h_in,
                 float* __restrict__ h_out,
                 float* __restrict__ cbuf,
                 float* __restrict__ out,
                 int s) {
  __shared__ float lds_h[2][B][KC + LPAD];  // 2 x 17 KB, TDM destination
  __shared__ float lds_g[4][B][16];         // 16 KB gate staging

  const int j0   = blockIdx.x * 16;   // hidden-column slice
  const int w    = threadIdx.x >> 5;
  const int lane = threadIdx.x & 31;
  const int gate = w >> 1;
  const int mt0  = (w & 1) * 2;       // first of two 16-row tiles
  const int l    = lane & 15;
  const int half = lane >> 4;

  const bool issuer = (w == 0);
  const unsigned lbase0 = lds_off(&lds_h[0][0][0]);
  const unsigned lbase1 = lds_off(&lds_h[1][0][0]);

  const float2* Brow = (const float2*)(w_hh + (size_t)(gate * H + j0 + l) * H);
  __builtin_prefetch(w_hh + (size_t)(gate * H + j0) * H, 0, 3);

  if (issuer) {
    tdm_load_h_chunk(h_in, 0, lbase0);
    __builtin_amdgcn_s_wait_tensorcnt(0);
  }
  __syncthreads();

  v8f acc0 = {};
  v8f acc1 = {};
  const int row0 = (mt0 * 16 + l) * (KC + LPAD);
  const int row1 = ((mt0 + 1) * 16 + l) * (KC + LPAD);

  for (int kc = 0; kc < NKC; ++kc) {
    if (issuer && (kc + 1) < NKC)
      tdm_load_h_chunk(h_in, (kc + 1) * KC, ((kc + 1) & 1) ? lbase1 : lbase0);

    const float* hs = &lds_h[kc & 1][0][0];
#pragma unroll
    for (int kk = 0; kk < KC; kk += 4) {
      int kg = kc * KC + kk;
      float2 bf = Brow[(kg >> 1) + half];
      v2f b; b.x = bf.x; b.y = bf.y;
      float2 a0f = *(const float2*)&hs[row0 + kk + half * 2];
      float2 a1f = *(const float2*)&hs[row1 + kk + half * 2];
      v2f a0; a0.x = a0f.x; a0.y = a0f.y;
      v2f a1; a1.x = a1f.x; a1.y = a1f.y;
      acc0 = wmma4(a0, b, acc0);
      acc1 = wmma4(a1, b, acc1);
    }

    if (issuer) __builtin_amdgcn_s_wait_tensorcnt(0);
    __syncthreads();
  }

#pragma unroll
  for (int r = 0; r < 8; ++r) {
    lds_g[gate][mt0 * 16 + r + half * 8][l]       = acc0[r];
    lds_g[gate][(mt0 + 1) * 16 + r + half * 8][l] = acc1[r];
  }
  __syncthreads();

  // Epilogue: 64 rows x 16 cols = 1024 elements over 256 threads.
#pragma unroll
  for (int e = threadIdx.x; e < B * 16; e += 256) {
    int row = e >> 4;        // batch index
    int col = e & 15;        // column within slice
    int j   = j0 + col;
    const float* xg = Xg + (size_t)(s * B + row) * G;
    float gi = lds_g[0][row][col] + xg[0 * H + j];
    float gf = lds_g[1][row][col] + xg[1 * H + j];
    float gg = lds_g[2][row][col] + xg[2 * H + j];
    float go = lds_g[3][row][col] + xg[3 * H + j];
    float cp = cbuf[(size_t)row * H + j];
    float cn = sigmoidf_(gf) * cp + sigmoidf_(gi) * tanhf(gg);
    float hn = sigmoidf_(go) * tanhf(cn);
    cbuf[(size_t)row * H + j]  = cn;
    h_out[(size_t)row * H + j] = hn;
    out[(size_t)row * S * H + (size_t)s * H + j] = hn;
  }
}

// ---------------------------------------------------------------------------
// Host-side launch
// ---------------------------------------------------------------------------
extern "C" void kernel_launch(void* const* d_in, const int* in_sizes, int n_in,
                              void* d_out, int out_size, void* d_ws, size_t ws_size,
                              hipStream_t stream) {
  const float* features = (const float*)d_in[0];
  const int*   captions = (const int*)  d_in[1];
  const float* W_embed  = (const float*)d_in[2];
  const float* b_embed  = (const float*)d_in[3];
  const float* w_ih     = (const float*)d_in[4];
  const float* w_hh     = (const float*)d_in[5];
  const float* b_ih     = (const float*)d_in[6];
  const float* b_hh     = (const float*)d_in[7];
  const float* h0       = (const float*)d_in[8];
  const float* c0       = (const float*)d_in[9];
  float* out = (float*)d_out;

  // Workspace layout (floats)
  float* ws = (float*)d_ws;
  const size_t X_OFF  = 0;
  const size_t XG_OFF = X_OFF + (size_t)S * B * E;     // 1,081,344
  const size_t H0_OFF = XG_OFF + (size_t)S * B * G;    // + 8,650,752
  const size_t H1_OFF = H0_OFF + (size_t)B * H;
  const size_t C_OFF  = H1_OFF + (size_t)B * H;
  float* X     = ws + X_OFF;
  float* Xg    = ws + XG_OFF;
  float* hbuf0 = ws + H0_OFF;
  float* hbuf1 = ws + H1_OFF;
  float* cbuf  = ws + C_OFF;

  // A: gather embeddings + init h/c
  {
    int n = S * B * E + 2 * B * H;
    int blocks = (n + 255) / 256;
    build_inputs_kernel<<<blocks, 256, 0, stream>>>(
        features, captions, W_embed, b_embed, h0, c0, X, hbuf0, cbuf);
  }

  // B: input projection for all timesteps at once
  {
    int tiles  = ((S * B) / 16) * (G / 16);  // 132 * 256
    int blocks = (tiles + 7) / 8;
    input_gemm_kernel<<<blocks, 256, 0, stream>>>(X, w_ih, b_ih, b_hh, Xg);
  }

  // C: sequential recurrence, double-buffered h
  for (int s = 0; s < S; ++s) {
    const float* h_in = (s & 1) ? hbuf1 : hbuf0;
    float*       h_o  = (s & 1) ? hbuf0 : hbuf1;
    lstm_step_kernel<<<H / 16, 256, 0, stream>>>(
        Xg, w_hh, h_in, h_o, cbuf, out, s);
  }
}